// RadialAverage_31782757990851
// MI455X (gfx1250) — compile-verified
//
#include <hip/hip_runtime.h>
#include <cfloat>

typedef __attribute__((ext_vector_type(2))) float v2f;
typedef __attribute__((ext_vector_type(8))) float v8f;

#define IMG   256
#define NPIX  65536
#define BATCH 64
#define VBINS 182
#define VPAD  192            // pad 182 -> 192 (12 WMMA tiles of 16)
#define CHUNKS 16            // 16-row chunks per image
#define SLOTS  65            // 64 images + 1 count slot
#define VTILES (VPAD / 16)   // 12

// ---------------------------------------------------------------------------
// Kernel 1: stream pixels, luma-convert, bin by integer radius into LDS,
// then dump this workgroup's private 192-bin partial vector to d_ws.
// Workgroups [0, 1024): (image, chunk) binning of 20*luma.
// Workgroups [1024, 1040): count-only binning (slot 64) -> nbin partials.
// ---------------------------------------------------------------------------
__global__ __launch_bounds__(256) void radial_bin_kernel(const float* __restrict__ x,
                                                         float* __restrict__ partial) {
    __shared__ float bins[VPAD];
    const int t  = threadIdx.x;
    const int wg = blockIdx.x;
    for (int v = t; v < VPAD; v += 256) bins[v] = 0.0f;
    __syncthreads();

    int slot, chunk;
    if (wg < BATCH * CHUNKS) {
        slot  = wg >> 4;
        chunk = wg & 15;
        const float* base = x + (size_t)slot * (3 * NPIX);
        const int row0 = chunk * 16;
        #pragma unroll
        for (int it = 0; it < 4; ++it) {
            const int q   = (it * 256 + t) * 4;     // pixel within 4096-pixel chunk
            const int col = q & 255;
            const int row = row0 + (q >> 8);
            const int idx = row * IMG + col;
            const float4 r4 = *(const float4*)(base + idx);
            const float4 g4 = *(const float4*)(base + NPIX + idx);
            const float4 b4 = *(const float4*)(base + 2 * NPIX + idx);
            const float dy  = (float)(row - 128);
            const float dy2 = dy * dy;
            const float rr[4] = {r4.x, r4.y, r4.z, r4.w};
            const float gg[4] = {g4.x, g4.y, g4.z, g4.w};
            const float bb[4] = {b4.x, b4.y, b4.z, b4.w};
            #pragma unroll
            for (int j = 0; j < 4; ++j) {
                const float dx  = (float)(col + j - 128);
                const int   rad = (int)sqrtf(fmaf(dx, dx, dy2));
                const float l   = 20.0f * fmaf(0.114f, bb[j],
                                       fmaf(0.587f, gg[j], 0.299f * rr[j]));
                atomicAdd(&bins[rad], l);           // ds_add_f32 (no-return)
            }
        }
    } else {
        slot  = BATCH;                               // count slot
        chunk = wg - BATCH * CHUNKS;
        const int row0 = chunk * 16;
        #pragma unroll
        for (int it = 0; it < 4; ++it) {
            const int q   = (it * 256 + t) * 4;
            const int col = q & 255;
            const int row = row0 + (q >> 8);
            const float dy  = (float)(row - 128);
            const float dy2 = dy * dy;
            #pragma unroll
            for (int j = 0; j < 4; ++j) {
                const float dx  = (float)(col + j - 128);
                const int   rad = (int)sqrtf(fmaf(dx, dx, dy2));
                atomicAdd(&bins[rad], 1.0f);
            }
        }
    }
    __syncthreads();
    float* dst = partial + ((size_t)slot * CHUNKS + chunk) * VPAD;
    for (int v = t; v < VPAD; v += 256) dst[v] = bins[v];
}

// ---------------------------------------------------------------------------
// Kernel 2: reduce 16 chunk-partials per slot with WMMA:
//   D = ones(16x4) x P(4x16) chained over K=16  ->  every row of D = sum_c P[c][v]
// One wave per (slot, 16-bin tile); 65*12 = 780 wave-jobs.
// ---------------------------------------------------------------------------
__global__ __launch_bounds__(256) void wmma_reduce_kernel(const float* __restrict__ partial,
                                                          float* __restrict__ reduced) {
    const int lane = threadIdx.x & 31;
    const int wave = threadIdx.x >> 5;
    const int job  = blockIdx.x * 8 + wave;
    if (job >= SLOTS * VTILES) return;
    const int slot = job / VTILES;
    const int vt   = job - slot * VTILES;
    const int half = lane >> 4;
    const int v    = vt * 16 + (lane & 15);
    const float* P = partial + (size_t)slot * (CHUNKS * VPAD);

    v2f a; a.x = 1.0f; a.y = 1.0f;    // all-ones A: layout-independent
    v8f c = {};

    #pragma unroll
    for (int kk = 0; kk < 4; ++kk) {
        const int c0 = kk * 4 + half * 2;            // k-assignment permutation is harmless
        v2f bfrag;
        bfrag.x = P[(c0 + 0) * VPAD + v];
        bfrag.y = P[(c0 + 1) * VPAD + v];
        c = __builtin_amdgcn_wmma_f32_16x16x4_f32(false, a, false, bfrag,
                                                  (short)0, c, false, false);
    }
    // D row 0: VGPR 0, lanes 0..15, N = lane
    if (lane < 16) reduced[slot * VPAD + v] = c[0];
}

// ---------------------------------------------------------------------------
// Kernel 3: profile = sums/nbin, global min-max normalize (single workgroup).
// ---------------------------------------------------------------------------
__global__ __launch_bounds__(256) void finalize_kernel(const float* __restrict__ reduced,
                                                       float* __restrict__ out) {
    __shared__ float smin[256];
    __shared__ float smax[256];
    const int t = threadIdx.x;
    const float* nbin = reduced + BATCH * VPAD;

    float lmin = FLT_MAX, lmax = -FLT_MAX;
    for (int i = t; i < BATCH * VBINS; i += 256) {
        const int b = i / VBINS;
        const int v = i - b * VBINS;
        const float p = reduced[b * VPAD + v] / nbin[v];
        lmin = fminf(lmin, p);
        lmax = fmaxf(lmax, p);
    }
    smin[t] = lmin; smax[t] = lmax;
    __syncthreads();
    for (int s = 128; s > 0; s >>= 1) {
        if (t < s) {
            smin[t] = fminf(smin[t], smin[t + s]);
            smax[t] = fmaxf(smax[t], smax[t + s]);
        }
        __syncthreads();
    }
    const float pmin = smin[0];
    const float inv  = 1.0f / (smax[0] - pmin);
    for (int i = t; i < BATCH * VBINS; i += 256) {
        const int b = i / VBINS;
        const int v = i - b * VBINS;
        const float p = reduced[b * VPAD + v] / nbin[v];
        out[i] = (p - pmin) * inv;
    }
}

extern "C" void kernel_launch(void* const* d_in, const int* in_sizes, int n_in,
                              void* d_out, int out_size, void* d_ws, size_t ws_size,
                              hipStream_t stream) {
    (void)in_sizes; (void)n_in; (void)out_size; (void)ws_size;
    const float* x   = (const float*)d_in[0];
    float* out       = (float*)d_out;
    float* partial   = (float*)d_ws;                               // 65*16*192 f32
    float* reduced   = partial + (size_t)SLOTS * CHUNKS * VPAD;    // 65*192 f32

    radial_bin_kernel<<<BATCH * CHUNKS + CHUNKS, 256, 0, stream>>>(x, partial);
    wmma_reduce_kernel<<<(SLOTS * VTILES + 7) / 8, 256, 0, stream>>>(partial, reduced);
    finalize_kernel<<<1, 256, 0, stream>>>(reduced, out);
}